// EdgeGNN_36481452212893
// MI455X (gfx1250) — compile-verified
//
#include <hip/hip_runtime.h>
#include <hip/hip_bf16.h>

#define NEG_SLOPE 0.2f

typedef __attribute__((ext_vector_type(2))) float v2f;
typedef __attribute__((ext_vector_type(8))) float v8f;

__device__ __forceinline__ unsigned fkey(float x) {
    unsigned u = __float_as_uint(x);
    return (u & 0x80000000u) ? ~u : (u | 0x80000000u);
}
__device__ __forceinline__ float fdekey(unsigned k) {
    return __uint_as_float((k & 0x80000000u) ? (k ^ 0x80000000u) : ~k);
}
__device__ __forceinline__ float lrelu(float x) { return x > 0.f ? x : NEG_SLOPE * x; }

// ---------------- wvec = We @ a_edge  ([32,64] @ [64]) ----------------
__global__ void k_wvec(const float* __restrict__ We, const float* __restrict__ ae,
                       float* __restrict__ wvec) {
    int k = threadIdx.x;
    if (k < 32) {
        float s = 0.f;
        for (int j = 0; j < 64; ++j) s += We[k * 64 + j] * ae[j];
        wvec[k] = s;
    }
}

// -------- per-edge score = edge_attr . wvec ; scsum[dst]+=s ; deg[dst]+=1 --------
__global__ void k_edge_score(const float* __restrict__ eattr, const int* __restrict__ dst,
                             const float* __restrict__ wvec, float* __restrict__ e_score,
                             float* __restrict__ scsum, float* __restrict__ deg,
                             int E, int addDeg) {
    int e = blockIdx.x * blockDim.x + threadIdx.x;
    if (e >= E) return;
    const float4* ea = (const float4*)(eattr + (size_t)e * 32);
    float s = 0.f;
#pragma unroll
    for (int i = 0; i < 8; ++i) {
        float4 v = ea[i];
        s += v.x * wvec[4 * i + 0] + v.y * wvec[4 * i + 1] +
             v.z * wvec[4 * i + 2] + v.w * wvec[4 * i + 3];
    }
    e_score[e] = s;
    int d = dst[e];
    atomicAdd(&scsum[d], s);
    if (addDeg) atomicAdd(&deg[d], 1.0f);
}

// ---------------- H = X @ W via V_WMMA_F32_16X16X4_F32 ----------------
// grid.x = ceil(Nrows/16), block = 128 (4 waves, one 16-wide N tile each; HID=64)
#define LDA_PAD 132   // K up to 128; 132 % 64 == 4 -> conflict-free A reads
__global__ void k_node_gemm(const float* __restrict__ X, const float* __restrict__ W,
                            float* __restrict__ Hout, int Nrows, int K) {
    __shared__ float lA[16 * LDA_PAD];
    __shared__ float lW[128 * 64];
    int m0 = blockIdx.x * 16;
    int tid = threadIdx.x;
    for (int i = tid; i < 16 * K; i += blockDim.x) {
        int r = i / K, c = i - r * K;
        int gr = m0 + r;
        lA[r * LDA_PAD + c] = (gr < Nrows) ? X[(size_t)gr * K + c] : 0.f;
    }
    for (int i = tid; i < K * 64; i += blockDim.x) lW[i] = W[i];
    __syncthreads();

    int lane = tid & 31;
    int ntile = tid >> 5;            // 0..3
    int mrow = lane & 15;
    int khalf = 2 * (lane >> 4);
    v8f acc = {};
    for (int kb = 0; kb < K; kb += 4) {
        v2f a, b;
        a.x = lA[mrow * LDA_PAD + kb + khalf];
        a.y = lA[mrow * LDA_PAD + kb + khalf + 1];
        b.x = lW[(kb + khalf) * 64 + ntile * 16 + mrow];
        b.y = lW[(kb + khalf + 1) * 64 + ntile * 16 + mrow];
        acc = __builtin_amdgcn_wmma_f32_16x16x4_f32(false, a, false, b,
                                                    (short)0, acc, false, false);
    }
    int col = ntile * 16 + mrow;
    int rbase = m0 + 8 * (lane >> 4);
#pragma unroll
    for (int r = 0; r < 8; ++r) {
        int gr = rbase + r;
        if (gr < Nrows) Hout[(size_t)gr * 64 + col] = acc[r];
    }
}

// ------- node: s_src, s_dst, self-loop logit, init segment-max key -------
__global__ void k_node_score(const float* __restrict__ h, const float* __restrict__ a_s,
                             const float* __restrict__ a_d, const float* __restrict__ scsum,
                             const float* __restrict__ deg, float* __restrict__ s_src,
                             float* __restrict__ s_dst, float* __restrict__ selfl,
                             unsigned* __restrict__ mkey, int N) {
    int i = blockIdx.x * blockDim.x + threadIdx.x;
    if (i >= N) return;
    const float4* hp = (const float4*)(h + (size_t)i * 64);
    float ss = 0.f, sd = 0.f;
#pragma unroll
    for (int q = 0; q < 16; ++q) {
        float4 v = hp[q];
        ss += v.x * a_s[4 * q] + v.y * a_s[4 * q + 1] + v.z * a_s[4 * q + 2] + v.w * a_s[4 * q + 3];
        sd += v.x * a_d[4 * q] + v.y * a_d[4 * q + 1] + v.z * a_d[4 * q + 2] + v.w * a_d[4 * q + 3];
    }
    s_src[i] = ss;
    s_dst[i] = sd;
    float hl = scsum[i] / fmaxf(deg[i], 1.0f);
    float lg = lrelu(ss + sd + hl);
    selfl[i] = lg;
    mkey[i] = fkey(lg);
}

// ------- edge pass A: logit (in-place over e_score) + segment max -------
__global__ void k_edge_logit(const int* __restrict__ src, const int* __restrict__ dst,
                             const float* __restrict__ s_src, const float* __restrict__ s_dst,
                             float* __restrict__ e_score, unsigned* __restrict__ mkey, int E) {
    int e = blockIdx.x * blockDim.x + threadIdx.x;
    if (e >= E) return;
    float lg = lrelu(s_src[src[e]] + s_dst[dst[e]] + e_score[e]);
    e_score[e] = lg;  // now holds logit
    atomicMax(&mkey[dst[e]], fkey(lg));
}

// ------- edge pass B: ex, denom, accum[dst] += ex*h[src] (4 thr/edge) -------
__global__ void k_edge_scatter(const int* __restrict__ src, const int* __restrict__ dst,
                               const float* __restrict__ logit, const unsigned* __restrict__ mkey,
                               const float* __restrict__ h, float* __restrict__ denom,
                               float* __restrict__ accum, int E) {
    int t = blockIdx.x * blockDim.x + threadIdx.x;
    int e = t >> 2, q = t & 3;
    if (e >= E) return;
    int d = dst[e], s = src[e];
    float ex = __expf(logit[e] - fdekey(mkey[d]));
    if (q == 0) atomicAdd(&denom[d], ex);
    const float4* hp = (const float4*)(h + (size_t)s * 64 + q * 16);
    float* ap = accum + (size_t)d * 64 + q * 16;
#pragma unroll
    for (int r = 0; r < 4; ++r) {
        float4 v = hp[r];
        atomicAdd(ap + 4 * r + 0, ex * v.x);
        atomicAdd(ap + 4 * r + 1, ex * v.y);
        atomicAdd(ap + 4 * r + 2, ex * v.z);
        atomicAdd(ap + 4 * r + 3, ex * v.w);
    }
}

// ------- node finalize: out = relu((accum + ex_self*h)/denom_tot + b) -------
__global__ void k_node_final(const float* __restrict__ h, const float* __restrict__ accum,
                             const float* __restrict__ denom, const float* __restrict__ selfl,
                             const unsigned* __restrict__ mkey, const float* __restrict__ bias,
                             float* __restrict__ xout, int N) {
    int i = blockIdx.x * blockDim.x + threadIdx.x;
    if (i >= N) return;
    float exs = __expf(selfl[i] - fdekey(mkey[i]));
    float inv = 1.0f / (denom[i] + exs);
    const float4* hp = (const float4*)(h + (size_t)i * 64);
    const float4* ap = (const float4*)(accum + (size_t)i * 64);
    const float4* bp = (const float4*)bias;
    float4* op = (float4*)(xout + (size_t)i * 64);
#pragma unroll
    for (int q = 0; q < 16; ++q) {
        float4 hv = hp[q], av = ap[q], bv = bp[q], o;
        o.x = fmaxf((av.x + exs * hv.x) * inv + bv.x, 0.f);
        o.y = fmaxf((av.y + exs * hv.y) * inv + bv.y, 0.f);
        o.z = fmaxf((av.z + exs * hv.z) * inv + bv.z, 0.f);
        o.w = fmaxf((av.w + exs * hv.w) * inv + bv.w, 0.f);
        op[q] = o;
    }
}

// ------- classifier: out[e] = relu([x2[s],x2[d],ea] @ Wc1 + bc1) @ Wc2 + bc2 -------
// grid.x = ceil(E/16), block = 128 (4 waves, 16-wide N tiles). K = 160, WMMA f32.
#define KC 160
#define LDC_PAD 164  // 164 % 64 == 36 -> distinct banks for 16 rows
__global__ void k_edge_mlp(const float* __restrict__ x2, const int* __restrict__ src,
                           const int* __restrict__ dst, const float* __restrict__ eattr,
                           const float* __restrict__ Wc1, const float* __restrict__ bc1,
                           const float* __restrict__ Wc2, const float* __restrict__ bc2,
                           float* __restrict__ out, int E) {
    __shared__ float lA[16 * LDC_PAD];
    __shared__ float lW[KC * 64];
    __shared__ float part[16];
    int e0 = blockIdx.x * 16;
    int tid = threadIdx.x;
    for (int i = tid; i < KC * 64; i += blockDim.x) lW[i] = Wc1[i];
    for (int i = tid; i < 16 * KC; i += blockDim.x) {
        int r = i / KC, c = i - r * KC;
        int e = e0 + r;
        float v = 0.f;
        if (e < E) {
            if (c < 64)       v = x2[(size_t)src[e] * 64 + c];
            else if (c < 128) v = x2[(size_t)dst[e] * 64 + (c - 64)];
            else              v = eattr[(size_t)e * 32 + (c - 128)];
        }
        lA[r * LDC_PAD + c] = v;
    }
    if (tid < 16) part[tid] = bc2[0];
    __syncthreads();

    int lane = tid & 31;
    int ntile = tid >> 5;
    int mrow = lane & 15;
    int khalf = 2 * (lane >> 4);
    v8f acc = {};
    for (int kb = 0; kb < KC; kb += 4) {
        v2f a, b;
        a.x = lA[mrow * LDC_PAD + kb + khalf];
        a.y = lA[mrow * LDC_PAD + kb + khalf + 1];
        b.x = lW[(kb + khalf) * 64 + ntile * 16 + mrow];
        b.y = lW[(kb + khalf + 1) * 64 + ntile * 16 + mrow];
        acc = __builtin_amdgcn_wmma_f32_16x16x4_f32(false, a, false, b,
                                                    (short)0, acc, false, false);
    }
    int col = ntile * 16 + mrow;
    float b1v = bc1[col], w2v = Wc2[col];
#pragma unroll
    for (int r = 0; r < 8; ++r) {
        float t = fmaxf(acc[r] + b1v, 0.f) * w2v;
        for (int off = 1; off < 16; off <<= 1) t += __shfl_xor(t, off, 16);
        if (mrow == 0) atomicAdd(&part[r + 8 * (lane >> 4)], t);
    }
    __syncthreads();
    if (tid < 16 && e0 + tid < E) out[e0 + tid] = part[tid];
}

// ============================== driver ==============================
static void run_gat_layer(const float* x, int K, const float* W, const float* We,
                          const float* a_s, const float* a_d, const float* a_e,
                          const float* bias, const int* src, const int* dst,
                          const float* eattr, int N, int E, int firstLayer,
                          float* h, float* accum, float* xout, float* s_src, float* s_dst,
                          float* deg, float* scsum, float* selfl, float* denom,
                          unsigned* mkey, float* wvec, float* e_score, hipStream_t stream) {
    hipMemsetAsync(scsum, 0, (size_t)N * 4, stream);
    hipMemsetAsync(denom, 0, (size_t)N * 4, stream);
    hipMemsetAsync(accum, 0, (size_t)N * 64 * 4, stream);
    if (firstLayer) hipMemsetAsync(deg, 0, (size_t)N * 4, stream);

    k_wvec<<<1, 32, 0, stream>>>(We, a_e, wvec);
    k_edge_score<<<(E + 255) / 256, 256, 0, stream>>>(eattr, dst, wvec, e_score, scsum, deg, E, firstLayer);
    k_node_gemm<<<(N + 15) / 16, 128, 0, stream>>>(x, W, h, N, K);
    k_node_score<<<(N + 255) / 256, 256, 0, stream>>>(h, a_s, a_d, scsum, deg, s_src, s_dst, selfl, mkey, N);
    k_edge_logit<<<(E + 255) / 256, 256, 0, stream>>>(src, dst, s_src, s_dst, e_score, mkey, E);
    k_edge_scatter<<<(4 * E + 255) / 256, 256, 0, stream>>>(src, dst, e_score, mkey, h, denom, accum, E);
    k_node_final<<<(N + 255) / 256, 256, 0, stream>>>(h, accum, denom, selfl, mkey, bias, xout, N);
}

extern "C" void kernel_launch(void* const* d_in, const int* in_sizes, int n_in,
                              void* d_out, int out_size, void* d_ws, size_t ws_size,
                              hipStream_t stream) {
    const float* x     = (const float*)d_in[0];
    const int*   ei    = (const int*)d_in[1];
    const float* eattr = (const float*)d_in[2];
    const float* W1  = (const float*)d_in[3];
    const float* We1 = (const float*)d_in[4];
    const float* as1 = (const float*)d_in[5];
    const float* ad1 = (const float*)d_in[6];
    const float* ae1 = (const float*)d_in[7];
    const float* b1  = (const float*)d_in[8];
    const float* W2  = (const float*)d_in[9];
    const float* We2 = (const float*)d_in[10];
    const float* as2 = (const float*)d_in[11];
    const float* ad2 = (const float*)d_in[12];
    const float* ae2 = (const float*)d_in[13];
    const float* b2  = (const float*)d_in[14];
    const float* Wc1 = (const float*)d_in[15];
    const float* bc1 = (const float*)d_in[16];
    const float* Wc2 = (const float*)d_in[17];
    const float* bc2 = (const float*)d_in[18];

    const int N = in_sizes[0] / 128;   // NODE_DIM = 128
    const int E = in_sizes[1] / 2;     // edge_index [2,E]
    const int* src = ei;
    const int* dst = ei + E;

    float* ws = (float*)d_ws;
    size_t NH = (size_t)N * 64;
    float* h      = ws;
    float* accum  = ws + NH;
    float* x1     = ws + 2 * NH;
    float* x2     = ws + 3 * NH;
    float* s_src  = ws + 4 * NH;
    float* s_dst  = s_src + N;
    float* deg    = s_dst + N;
    float* scsum  = deg + N;
    float* selfl  = scsum + N;
    float* denom  = selfl + N;
    unsigned* mkey = (unsigned*)(denom + N);
    float* wvec   = (float*)(mkey + N);
    float* e_score = wvec + 32;        // doubles as logit buffer (in-place)

    // Layer 1: input x [N,128]
    run_gat_layer(x, 128, W1, We1, as1, ad1, ae1, b1, src, dst, eattr, N, E, 1,
                  h, accum, x1, s_src, s_dst, deg, scsum, selfl, denom, mkey, wvec,
                  e_score, stream);
    // Layer 2: input x1 [N,64]
    run_gat_layer(x1, 64, W2, We2, as2, ad2, ae2, b2, src, dst, eattr, N, E, 0,
                  h, accum, x2, s_src, s_dst, deg, scsum, selfl, denom, mkey, wvec,
                  e_score, stream);
    // Edge classifier
    k_edge_mlp<<<(E + 15) / 16, 128, 0, stream>>>(x2, src, dst, eattr, Wc1, bc1, Wc2, bc2,
                                                  (float*)d_out, E);
}